// MaxPoolFaceFeature_43748536877374
// MI455X (gfx1250) — compile-verified
//
#include <hip/hip_runtime.h>
#include <cstdint>

// Problem constants (from reference)
#define NUM_MESHES   8
#define IN_CHANNEL   256
#define NUM_FACES    20000
#define NUM_NEIGHBOR 3

// Channels staged per block: 2 rows * 20000 f32 = 160000 B LDS -> 2 blocks/WGP (320KB)
#define CHUNK        2
#define BLOCK        256
#define LDS_BYTES    (CHUNK * NUM_FACES * 4)

// out[m,c,f] = max(fea[m,c,f], max_k fea[m,c,ring[m,f,k]])
//
// Strategy: one block owns (m, c0..c0+CHUNK-1). Stage the CHUNK rows fea[m,c,:]
// (80KB each) into LDS with CDNA5 async global->LDS b128 copies (NT hint: each
// row is read exactly once, so it must not evict the L2-resident ring_n), then
// all random gathers become ds_load ops -- converting 123M scattered 4B L2
// requests into LDS bank traffic, which is what keeps us on the 23.3 TB/s
// HBM roofline. Output written with non-temporal stores for the same reason.
__global__ void MaxPoolFaceFeature_43748536877374_kernel(
    const float* __restrict__ fea,   // [M, C, F]
    const int*   __restrict__ ring,  // [M, F, K] (harness passes integer inputs as int32)
    float*       __restrict__ out)   // [M, C, F]
{
    extern __shared__ float rows[];  // CHUNK * NUM_FACES floats

    const int tid = threadIdx.x;
    const int m   = blockIdx.y;
    const int c0  = blockIdx.x * CHUNK;

    // ---- Stage CHUNK channel rows into LDS via async global->LDS copies ----
    // LDS byte offset = low 32 bits of the generic shared-memory address
    // (flat LDS aperture keeps the offset in addr[31:0]).
    const uint32_t lds_base = (uint32_t)(uintptr_t)rows;
    const float*   grow0    = fea + ((size_t)m * IN_CHANNEL + c0) * NUM_FACES;

    #pragma unroll
    for (int r = 0; r < CHUNK; ++r) {
        const float* grow = grow0 + (size_t)r * NUM_FACES;
        const uint32_t lrow = lds_base + (uint32_t)(r * NUM_FACES * 4);
        // 20000 floats = 5000 x b128 transfers per row, spread over 256 threads.
        for (int d = tid; d < NUM_FACES / 4; d += BLOCK) {
            uint64_t gaddr = (uint64_t)(uintptr_t)(grow + 4 * d);
            uint32_t laddr = lrow + (uint32_t)(16 * d);
            // NT: fea is single-use streaming data; keep L2 for ring_n.
            asm volatile("global_load_async_to_lds_b128 %0, %1, off th:TH_LOAD_NT"
                         :: "v"(laddr), "v"(gaddr)
                         : "memory");
        }
    }
    // Wait for this wave's async copies, then barrier so all waves see full LDS.
    asm volatile("s_wait_asynccnt 0" ::: "memory");
    __syncthreads();

    // ---- Gather-max: all random accesses now hit LDS (64 banks, wave32) ----
    const int*   rrow    = ring + (size_t)m * NUM_FACES * NUM_NEIGHBOR;
    const size_t outbase = ((size_t)m * IN_CHANNEL + c0) * NUM_FACES;

    // Unroll x2: with 16 waves/WGP, extra in-flight ds_loads + index loads
    // per wave are needed to cover LDS/L2 latency.
    #pragma unroll 2
    for (int f = tid; f < NUM_FACES; f += BLOCK) {
        // Streaming, coalesced index read (L2-resident, reused for CHUNK channels).
        const int i0 = rrow[3 * f + 0];
        const int i1 = rrow[3 * f + 1];
        const int i2 = rrow[3 * f + 2];

        #pragma unroll
        for (int r = 0; r < CHUNK; ++r) {
            const float* L = rows + r * NUM_FACES;
            float v = L[f];                 // self
            v = fmaxf(v, L[i0]);            // ds_load gathers
            v = fmaxf(v, L[i1]);
            v = fmaxf(v, L[i2]);
            // NT store: don't pollute L2 with the 164MB output stream.
            __builtin_nontemporal_store(v, out + outbase + (size_t)r * NUM_FACES + f);
        }
    }
}

extern "C" void kernel_launch(void* const* d_in, const int* in_sizes, int n_in,
                              void* d_out, int out_size, void* d_ws, size_t ws_size,
                              hipStream_t stream) {
    const float* fea  = (const float*)d_in[0];
    const int*   ring = (const int*)d_in[1];
    float*       out  = (float*)d_out;

    // Allow 160KB dynamic LDS per block (gfx1250 WGP has 320KB).
    (void)hipFuncSetAttribute(
        (const void*)MaxPoolFaceFeature_43748536877374_kernel,
        hipFuncAttributeMaxDynamicSharedMemorySize, LDS_BYTES);

    dim3 grid(IN_CHANNEL / CHUNK, NUM_MESHES);  // (128, 8) blocks
    MaxPoolFaceFeature_43748536877374_kernel<<<grid, BLOCK, LDS_BYTES, stream>>>(
        fea, ring, out);
}